// CrossInvolution_55035710931681
// MI455X (gfx1250) — compile-verified
//
#include <hip/hip_runtime.h>

// Problem constants (match reference)
static constexpr int Bn   = 2;
static constexpr int Cc   = 256;
static constexpr int Hh   = 64;
static constexpr int Ww   = 64;
static constexpr int HW   = Hh * Ww;      // 4096
static constexpr int REDc = 64;
static constexpr int Gg   = 16;           // groups
static constexpr int GCc  = 16;           // group channels
static constexpr int K7   = 7;
static constexpr int KK   = 49;
static constexpr int OC2  = KK * Gg;      // 784
static constexpr float EPSf = 1e-5f;

typedef float v2f __attribute__((ext_vector_type(2)));
typedef float v8f __attribute__((ext_vector_type(8)));

#if defined(__gfx1250__) && __has_builtin(__builtin_amdgcn_global_load_async_to_lds_b32)
#define USE_ASYNC_LDS 1
typedef __attribute__((address_space(1))) int* gas_ptr;   // global (AS1) int*
typedef __attribute__((address_space(3))) int* las_ptr;   // LDS (AS3) int*
#else
#define USE_ASYNC_LDS 0
#endif

// ---------------------------------------------------------------------------
// Tiny prep: fold BN into per-channel scale/shift
// ---------------------------------------------------------------------------
__global__ void bn_prep_kernel(const float* __restrict__ gamma,
                               const float* __restrict__ beta,
                               const float* __restrict__ mean,
                               const float* __restrict__ var,
                               float* __restrict__ scale,
                               float* __restrict__ shift) {
    int i = threadIdx.x;
    if (i < REDc) {
        float s = gamma[i] * rsqrtf(var[i] + EPSf);
        scale[i] = s;
        shift[i] = beta[i] - mean[i] * s;
    }
}

// ---------------------------------------------------------------------------
// conv1 as fp32 WMMA GEMM: t[b] = ReLU(BN(w1 (64x256) @ guide[b] (256x4096)))
// One wave -> one 16x16 tile. K stepped by 4 via v_wmma_f32_16x16x4_f32.
// ---------------------------------------------------------------------------
__global__ __launch_bounds__(256)
void conv1_wmma_f32(const float* __restrict__ A,      // w1: 64 x 256
                    const float* __restrict__ Bmat,   // guide: (b) 256 x 4096
                    float* __restrict__ Cmat,         // t: (b) 64 x 4096
                    const float* __restrict__ scale,
                    const float* __restrict__ shift) {
    const int bb = blockIdx.y;
    const float* Bb = Bmat + (size_t)bb * Cc * HW;
    float*       Cb = Cmat + (size_t)bb * REDc * HW;

    const int waveId = blockIdx.x * (blockDim.x >> 5) + (threadIdx.x >> 5);
    const int tilesN = HW >> 4;                  // 256
    if (waveId >= (REDc >> 4) * tilesN) return;  // wave-uniform guard

    const int tm = waveId / tilesN;
    const int tn = waveId % tilesN;
    const int lane = threadIdx.x & 31;
    const int lo = lane & 15;
    const int hi = lane >> 4;

    const float* arow = A  + (size_t)(tm * 16 + lo) * Cc;  // A: lane lo -> M=lo
    const float* bcol = Bb + (size_t)(tn * 16 + lo);       // B: lane lo -> N=lo

    v8f acc = {};
    for (int k = 0; k < Cc; k += 4) {
        const int ka = k + 2 * hi;                          // K pair per lane-half
        v2f afrag; afrag.x = arow[ka]; afrag.y = arow[ka + 1];
        const float* bp = bcol + (size_t)ka * HW;
        v2f bfrag; bfrag.x = bp[0]; bfrag.y = bp[HW];
        acc = __builtin_amdgcn_wmma_f32_16x16x4_f32(
            false, afrag, false, bfrag, (short)0, acc, false, false);
    }

    const int col = tn * 16 + lo;
#pragma unroll
    for (int r = 0; r < 8; ++r) {
        const int row = tm * 16 + r + 8 * hi;   // C/D: VGPR r, hi half -> M=8+r
        float v = acc[r] * scale[row] + shift[row];
        v = v > 0.f ? v : 0.f;                   // ReLU
        Cb[(size_t)row * HW + col] = v;
    }
}

// ---------------------------------------------------------------------------
// Fused conv2 + involution + residual.
// Block = 16x16 pixel tile for one (batch, group).
//   Phase 0: feature halo tile (16ch x 22x22) -> LDS, via async DMA
//            (GLOBAL_LOAD_ASYNC_TO_LDS_B32) so it overlaps Phase 1.
//   Phase 1: weight tile = w2[g*49:(g+1)*49] @ t[:, tile] + b2  via WMMA -> LDS
//            (M=49 padded to 4x16 tiles; N = the 16 pixel rows; K=64)
//   Phase 2: 49-tap stencil per pixel, broadcast over 16 group channels.
// The weight tensor (2 x 25.7 MB round-trip in the naive version) never
// touches HBM; t (2 MB) is L2-resident across its 16x group re-reads.
// ---------------------------------------------------------------------------
static constexpr int TS   = 16;
static constexpr int HALO = 3;
static constexpr int SW   = TS + 2 * HALO;   // 22
static constexpr int WST  = 260;             // weight-LDS row stride: 8*260 % 64 == 32
                                             // -> the two lane-halves hit disjoint banks

__global__ __launch_bounds__(256)
void fused_conv2_involution(const float* __restrict__ feat,
                            const float* __restrict__ tbuf,  // (b) 64 x 4096
                            const float* __restrict__ w2,    // 784 x 64
                            const float* __restrict__ b2,    // 784
                            float* __restrict__ out) {
    __shared__ float wLDS[KK][WST];             // 49 x 260 floats = 50,960 B
    __shared__ float sm[GCc][SW][SW];           // 16 x 22 x 22 floats = 30,976 B

    const int bz = blockIdx.z;
    const int bb = bz / Gg;
    const int g  = bz % Gg;
    const int x0 = blockIdx.x * TS;
    const int y0 = blockIdx.y * TS;
    const int tid = threadIdx.x;                // 0..255
    const int tx = tid & 15;
    const int ty = tid >> 4;

    // ---- Phase 0: feature halo-tile -> LDS (zero pad off-image) ----
    // In-bounds slots: async DMA global->LDS (ASYNCcnt). OOB slots: ds_store 0.
    // Disjoint address sets, so no DS-vs-async ordering hazard; both are
    // fenced by s_wait_asynccnt + __syncthreads() before the stencil reads.
    const float* fbase = feat + (size_t)(bb * Cc + g * GCc) * HW;
    for (int idx = tid; idx < GCc * SW * SW; idx += 256) {
        const int c   = idx / (SW * SW);
        const int rem = idx % (SW * SW);
        const int yy  = rem / SW;
        const int xx  = rem % SW;
        const int gy  = y0 + yy - HALO;
        const int gx  = x0 + xx - HALO;
        const bool inb = (gy >= 0) & (gy < Hh) & (gx >= 0) & (gx < Ww);
#if USE_ASYNC_LDS
        if (inb) {
            __builtin_amdgcn_global_load_async_to_lds_b32(
                (gas_ptr)(fbase + (size_t)c * HW + gy * Ww + gx),
                (las_ptr)&sm[c][yy][xx], 0, 0);
        } else {
            sm[c][yy][xx] = 0.f;
        }
#else
        float v = 0.f;
        if (inb) v = fbase[(size_t)c * HW + gy * Ww + gx];
        sm[c][yy][xx] = v;
#endif
    }

    // ---- Phase 1: WMMA weight-tile GEMM into LDS (overlaps async copies) ----
    const int lane = tid & 31;
    const int wave = tid >> 5;
    const int lo = lane & 15;
    const int hi = lane >> 4;
    const float* tB = tbuf + (size_t)bb * REDc * HW;

    for (int tile = wave; tile < 64; tile += 8) {       // wave-uniform loop
        const int tm = tile >> 4;                       // 0..3 (M sub-tile)
        const int tn = tile & 15;                       // pixel row of the tile (N)
        int arow_idx = g * KK + tm * 16 + lo;           // output-channel row
        if (arow_idx > OC2 - 1) arow_idx = OC2 - 1;     // clamp (g==15 overrun);
                                                        // results discarded below
        const float* arow = w2 + (size_t)arow_idx * REDc;
        const int colbase = (y0 + tn) * Ww + x0 + lo;   // pixel column in t

        v8f acc = {};
        for (int k = 0; k < REDc; k += 4) {
            const int ka = k + 2 * hi;
            v2f afrag; afrag.x = arow[ka]; afrag.y = arow[ka + 1];
            const float* bp = tB + (size_t)ka * HW + colbase;
            v2f bfrag; bfrag.x = bp[0]; bfrag.y = bp[HW];
            acc = __builtin_amdgcn_wmma_f32_16x16x4_f32(
                false, afrag, false, bfrag, (short)0, acc, false, false);
        }
#pragma unroll
        for (int r = 0; r < 8; ++r) {
            const int m = tm * 16 + r + 8 * hi;         // local weight row
            if (m < KK)                                  // drop padding rows
                wLDS[m][tn * 16 + lo] = acc[r] + b2[g * KK + m];
        }
    }

#if USE_ASYNC_LDS
    asm volatile("s_wait_asynccnt 0x0" ::: "memory");   // halo DMA complete
#endif
    __syncthreads();

    // ---- Phase 2: stencil (49 taps) over 16 group channels + residual ----
    float wreg[KK];
#pragma unroll
    for (int k = 0; k < KK; ++k) wreg[k] = wLDS[k][ty * 16 + tx];

    const int y = y0 + ty, x = x0 + tx;
    float* obase = out + (size_t)(bb * Cc + g * GCc) * HW + y * Ww + x;
    for (int c = 0; c < GCc; ++c) {
        float acc = sm[c][ty + HALO][tx + HALO];        // residual term
#pragma unroll
        for (int kh = 0; kh < K7; ++kh)
#pragma unroll
            for (int kw = 0; kw < K7; ++kw)
                acc = fmaf(wreg[kh * K7 + kw], sm[c][ty + kh][tx + kw], acc);
        obase[(size_t)c * HW] = acc;
    }
}

// ---------------------------------------------------------------------------
// Launch
// ---------------------------------------------------------------------------
extern "C" void kernel_launch(void* const* d_in, const int* in_sizes, int n_in,
                              void* d_out, int out_size, void* d_ws, size_t ws_size,
                              hipStream_t stream) {
    const float* feature = (const float*)d_in[0];
    const float* guide   = (const float*)d_in[1];
    const float* w1      = (const float*)d_in[2];
    const float* gamma   = (const float*)d_in[3];
    const float* beta    = (const float*)d_in[4];
    const float* mean    = (const float*)d_in[5];
    const float* var     = (const float*)d_in[6];
    const float* w2      = (const float*)d_in[7];
    const float* b2      = (const float*)d_in[8];
    float* out = (float*)d_out;

    char* ws = (char*)d_ws;
    float* scale = (float*)ws;             // 64 floats
    float* shift = (float*)(ws + 256);     // 64 floats
    float* tbuf  = (float*)(ws + 512);     // 2 * 64 * 4096 floats = 2 MB

    bn_prep_kernel<<<dim3(1), dim3(64), 0, stream>>>(gamma, beta, mean, var, scale, shift);

    // conv1: t = ReLU(BN(w1 @ guide))   M=64, N=4096, K=256 per batch
    {
        const int tiles = (REDc / 16) * (HW / 16);   // 1024 per batch
        dim3 grid((tiles + 7) / 8, Bn);
        conv1_wmma_f32<<<grid, 256, 0, stream>>>(w1, guide, tbuf, scale, shift);
    }

    // fused conv2 + involution + residual (weight tensor never touches HBM)
    fused_conv2_involution<<<dim3(Ww / TS, Hh / TS, Bn * Gg), dim3(256), 0, stream>>>(
        feature, tbuf, w2, b2, out);
}